// HybridBoundaryLoss_58162447122912
// MI455X (gfx1250) — compile-verified
//
#include <hip/hip_runtime.h>

// ---------------------------------------------------------------------------
// HybridBoundaryLoss for MI455X (gfx1250, wave32).
// pred: [8,1,384,384] f32, target: [8,1,384,384] i32 -> scalar f32 loss.
//
// loss2's [B,1,H,W]*[B,H,W] broadcast factorizes:
//   sum(multip) = sum_{h,w} Asum(h,w)*Ssum(h,w),  Asum=sum_b|p-t|, Ssum=sum_b|sdf|
// so we never materialize the [8,8,384,384] tensor.
//
// K1: pointwise BCE/dice/Asum, deterministic per-block tree partials
// K2: EDT column pass (fwd+bwd scan per column), stores squared distances
// K3: EDT row pass (min-plus lower envelope, brute force O(W^2) via LDS
//     broadcast) fused with |sdf|*Asum row reduction
// K4: single-wave final reduction using V_WMMA_F32_16X16X4_F32 with an
//     all-ones B matrix: D[m,n] = sum_k A[m,k] + C[m,n]  (fp32-exact sum of
//     64 values per instruction, chained through the C accumulator)
// ---------------------------------------------------------------------------

typedef __attribute__((ext_vector_type(2))) float v2f;
typedef __attribute__((ext_vector_type(8))) float v8f;

#define BB 8
#define HH 384
#define WW 384
#define HWSZ (HH * WW)          // 147456
#define NTOT (BB * HWSZ)        // 1179648
#define INF_ 1.0e4f
#define K1_BLOCKS (HWSZ / 256)  // 576  (= 9*64, WMMA-chunk friendly)
#define NPART 25                // bce + 8*(p*t) + 8*(p) + 8*(t)

// workspace layout (float offsets)
#define OFF_FPOS 0                               // [B,H,W] d1^2 for posdis
#define OFF_FNEG (NTOT)                          // [B,H,W] d1^2 for negdis
#define OFF_ASUM (2 * NTOT)                      // [H,W]   sum_b |p-t|
#define OFF_ROWP (2 * NTOT + HWSZ)               // [B*H]   row partial products (3072 = 48*64)
#define OFF_K1P  (2 * NTOT + HWSZ + BB * HH)     // [25][576] pointwise partials

// ------------------------- K1: pointwise --------------------------------
__global__ void k1_pointwise(const float* __restrict__ pred,
                             const int* __restrict__ target,
                             float* __restrict__ ws) {
  const int tid = threadIdx.x;
  const int idx = blockIdx.x * 256 + tid;  // pixel in [0, HW)
  float vals[NPART];
#pragma unroll
  for (int c = 0; c < NPART; ++c) vals[c] = 0.0f;
  float asum = 0.0f;
#pragma unroll
  for (int b = 0; b < BB; ++b) {
    const float x = pred[b * HWSZ + idx];
    const float t = (float)target[b * HWSZ + idx];
    const float ax = fabsf(x);
    const float sp = fmaxf(x, 0.0f) + log1pf(expf(-ax));  // stable softplus
    vals[0] += sp - x * t;
    const float p = 1.0f / (1.0f + expf(-x));
    vals[1 + b]  += p * t;
    vals[9 + b]  += p;
    vals[17 + b] += t;
    asum += fabsf(p - t);
  }
  ws[OFF_ASUM + idx] = asum;

  __shared__ float red[256];
  float* k1p = ws + OFF_K1P;
  for (int c = 0; c < NPART; ++c) {
    red[tid] = vals[c];
    __syncthreads();
    for (int s = 128; s > 0; s >>= 1) {
      if (tid < s) red[tid] += red[tid + s];
      __syncthreads();
    }
    if (tid == 0) k1p[c * K1_BLOCKS + blockIdx.x] = red[0];
    __syncthreads();
  }
}

// --------------------- K2: EDT column scans -----------------------------
__global__ void k2_coldist(const int* __restrict__ target, float* __restrict__ ws) {
  const int g = blockIdx.x * 256 + threadIdx.x;  // [0, B*W)
  if (g >= BB * WW) return;
  const int b = g / WW, x = g % WW;
  const int* col = target + b * HWSZ + x;
  float* fp = ws + OFF_FPOS + b * HWSZ + x;
  float* fn = ws + OFF_FNEG + b * HWSZ + x;

  float dp = INF_, dn = INF_;
  for (int y = 0; y < HH; ++y) {
    const bool pos = col[y * WW] != 0;
    dp = pos ? 0.0f : fminf(dp + 1.0f, INF_);
    dn = pos ? fminf(dn + 1.0f, INF_) : 0.0f;
    fp[y * WW] = dp;
    fn[y * WW] = dn;
  }
  dp = INF_; dn = INF_;
  for (int y = HH - 1; y >= 0; --y) {
    const bool pos = col[y * WW] != 0;
    dp = pos ? 0.0f : fminf(dp + 1.0f, INF_);
    dn = pos ? fminf(dn + 1.0f, INF_) : 0.0f;
    const float a = fminf(fp[y * WW], dp);
    const float c = fminf(fn[y * WW], dn);
    fp[y * WW] = a * a;   // squared column distance
    fn[y * WW] = c * c;
  }
}

// ----------- K3: row min-plus envelope + |sdf|*Asum reduce --------------
__global__ void k3_rowpass(float* ws) {
  const int row = blockIdx.x;        // [0, B*H)
  const int b = row / HH, y = row % HH;
  const int i = threadIdx.x;         // [0, W)

  __shared__ float gp[WW];
  __shared__ float gn[WW];
  __shared__ float red[WW];

  const float* fp = ws + OFF_FPOS + (b * HWSZ + y * WW);
  const float* fn = ws + OFF_FNEG + (b * HWSZ + y * WW);
  const float jf = (float)i;
  gp[i] = fp[i] + jf * jf;           // f[j] + j^2
  gn[i] = fn[i] + jf * jf;
  __syncthreads();

  const float fi = (float)i;
  const float two_i = 2.0f * fi;
  float mp = 3.0e38f, mn = 3.0e38f;
#pragma unroll 4
  for (int j = 0; j < WW; ++j) {
    const float fj = (float)j;
    mp = fminf(mp, fmaf(-two_i, fj, gp[j]));   // (f+j^2) - 2ij
    mn = fminf(mn, fmaf(-two_i, fj, gn[j]));
  }
  mp += fi * fi;                     // + i^2  -> min_j f[j]+(i-j)^2
  mn += fi * fi;
  const float sdf_abs = fabsf(sqrtf(mp) - sqrtf(mn));
  const float prod = sdf_abs * ws[OFF_ASUM + y * WW + i];

  red[i] = prod;
  __syncthreads();
  for (int s = 256; s > 0; s >>= 1) {
    if (i < s && i + s < WW) red[i] += red[i + s];
    __syncthreads();
  }
  if (i == 0) ws[OFF_ROWP + row] = red[0];
}

// ------------------- K4: WMMA-based final reduction ---------------------
// Sum of n = 64*nchunks floats with V_WMMA_F32_16X16X4_F32:
// A holds 64 values (2 VGPRs/lane * 32 lanes), B = ones, C chains chunks.
// D[m,n] = sum_k A[m,k] + C[m,n]  (independent of n) =>
// total = (sum over all 256 D slots) / 16.
__device__ __forceinline__ float wave_wmma_sum(const float* data, int nchunks, int lane) {
  v8f acc = {};
  v2f ones; ones.x = 1.0f; ones.y = 1.0f;
  for (int c = 0; c < nchunks; ++c) {
    v2f a;
    a.x = data[c * 64 + lane];
    a.y = data[c * 64 + 32 + lane];
    acc = __builtin_amdgcn_wmma_f32_16x16x4_f32(
        /*neg_a=*/false, a, /*neg_b=*/false, ones,
        /*c_mod=*/(short)0, acc, /*reuse_a=*/false, /*reuse_b=*/false);
  }
  float s = acc[0] + acc[1] + acc[2] + acc[3] + acc[4] + acc[5] + acc[6] + acc[7];
#pragma unroll
  for (int m = 16; m >= 1; m >>= 1) s += __shfl_xor(s, m, 32);
  return s * 0.0625f;  // /16: each row-sum is replicated across 16 columns
}

__global__ void k4_final(float* __restrict__ out, const float* __restrict__ ws) {
  const int lane = threadIdx.x;  // blockDim.x == 32: one full wave, EXEC all 1s
  const float* k1p = ws + OFF_K1P;

  float sums[NPART];
  for (int c = 0; c < NPART; ++c)
    sums[c] = wave_wmma_sum(k1p + c * K1_BLOCKS, K1_BLOCKS / 64, lane);  // 9 chunks
  const float prodsum = wave_wmma_sum(ws + OFF_ROWP, (BB * HH) / 64, lane);  // 48 chunks

  if (lane == 0) {
    const float bce = sums[0] / (float)NTOT;
    float dm = 0.0f;
#pragma unroll
    for (int b = 0; b < BB; ++b) {
      const float num = 2.0f * sums[1 + b] + 1e-5f;
      const float den = sums[9 + b] + sums[17 + b] + 1e-5f;
      dm += num / den;
    }
    dm *= (1.0f / (float)BB);
    const float loss1 = 0.5f * bce + (1.0f - dm);
    const float loss2 = prodsum / ((float)(BB * BB) * (float)HWSZ);
    out[0] = 0.7f * loss1 + 0.03f * loss2;
  }
}

// ---------------------------------------------------------------------------
extern "C" void kernel_launch(void* const* d_in, const int* in_sizes, int n_in,
                              void* d_out, int out_size, void* d_ws, size_t ws_size,
                              hipStream_t stream) {
  (void)in_sizes; (void)n_in; (void)out_size; (void)ws_size;
  const float* pred = (const float*)d_in[0];
  const int* target = (const int*)d_in[1];
  float* ws = (float*)d_ws;    // needs ~10.1 MB
  float* out = (float*)d_out;

  k1_pointwise<<<K1_BLOCKS, 256, 0, stream>>>(pred, target, ws);
  k2_coldist<<<(BB * WW + 255) / 256, 256, 0, stream>>>(target, ws);
  k3_rowpass<<<BB * HH, WW, 0, stream>>>(ws);
  k4_final<<<1, 32, 0, stream>>>(out, ws);
}